// QuantizedRMSELoss_9543417331713
// MI455X (gfx1250) — compile-verified
//
#include <hip/hip_runtime.h>
#include <hip/hip_bf16.h>
#include <math.h>

// ---------------------------------------------------------------------------
// QuantizedRMSELoss for MI455X (gfx1250, wave32).
// Memory-bound streaming reduction: 283 MB reads -> ~12.2 us floor @ 23.3 TB/s.
// Cumulative-threshold binning keeps VALU at ~22 ops/elem (cmp/cndmask/fmac).
// Deterministic two-kernel reduction (no atomics -> identical across graph
// replays). Final 4-term weighted-MSE dot product runs on the matrix unit via
// V_WMMA_F32_16X16X4_F32 (K=4 matches the dot length exactly, full f32).
// ---------------------------------------------------------------------------

typedef __attribute__((ext_vector_type(2))) float v2f;
typedef __attribute__((ext_vector_type(8))) float v8f;

#define NB 2048
#define NT 256
#define NWAVE (NT / 32)

__device__ __forceinline__ void accum_elem(float yp, float yt, float s[5], float c[5]) {
    // thresholds = float32(log1p([0,5,25,50,100]))
    const float T1 = 1.791759469228055f;   // log1p(5)
    const float T2 = 3.258096538021482f;   // log1p(25)
    const float T3 = 3.931825632724312f;   // log1p(50)
    const float T4 = 4.615120516841259f;   // log1p(100)
    float d  = yp - yt;
    float sq = d * d;
    float m0 = (yt >= 0.0f) ? 1.0f : 0.0f;
    float m1 = (yt >= T1)   ? 1.0f : 0.0f;
    float m2 = (yt >= T2)   ? 1.0f : 0.0f;
    float m3 = (yt >= T3)   ? 1.0f : 0.0f;
    float m4 = (yt >= T4)   ? 1.0f : 0.0f;
    s[0] = fmaf(m0, sq, s[0]);  c[0] += m0;
    s[1] = fmaf(m1, sq, s[1]);  c[1] += m1;
    s[2] = fmaf(m2, sq, s[2]);  c[2] += m2;
    s[3] = fmaf(m3, sq, s[3]);  c[3] += m3;
    s[4] = fmaf(m4, sq, s[4]);  c[4] += m4;
}

__global__ void __launch_bounds__(NT)
qrmse_partial(const float* __restrict__ yp, const float* __restrict__ yt,
              float* __restrict__ partials, int n) {
    float s[5] = {0.f, 0.f, 0.f, 0.f, 0.f};
    float c[5] = {0.f, 0.f, 0.f, 0.f, 0.f};

    const int n4     = n >> 2;
    const int gid    = blockIdx.x * blockDim.x + threadIdx.x;
    const int stride = gridDim.x * blockDim.x;

    const float4* __restrict__ yp4 = (const float4*)yp;
    const float4* __restrict__ yt4 = (const float4*)yt;

    // main vectorized grid-stride loop: two b128 loads per iteration
    for (int i = gid; i < n4; i += stride) {
        float4 a = yp4[i];
        float4 b = yt4[i];
        accum_elem(a.x, b.x, s, c);
        accum_elem(a.y, b.y, s, c);
        accum_elem(a.z, b.z, s, c);
        accum_elem(a.w, b.w, s, c);
    }
    // scalar tail (n not a multiple of 4)
    for (int i = (n4 << 2) + gid; i < n; i += stride) {
        accum_elem(yp[i], yt[i], s, c);
    }

    // wave32 butterfly reduction
#pragma unroll
    for (int off = 16; off > 0; off >>= 1) {
#pragma unroll
        for (int j = 0; j < 5; ++j) {
            s[j] += __shfl_xor(s[j], off, 32);
            c[j] += __shfl_xor(c[j], off, 32);
        }
    }

    __shared__ float red[NWAVE][10];
    const int lane = threadIdx.x & 31;
    const int wv   = threadIdx.x >> 5;
    if (lane == 0) {
#pragma unroll
        for (int j = 0; j < 5; ++j) {
            red[wv][j]     = s[j];
            red[wv][5 + j] = c[j];
        }
    }
    __syncthreads();

    if (threadIdx.x == 0) {
#pragma unroll
        for (int j = 0; j < 10; ++j) {
            float t = 0.f;
            for (int w = 0; w < NWAVE; ++w) t += red[w][j];
            partials[blockIdx.x * 10 + j] = t;   // unconditional -> no ws init needed
        }
    }
}

__global__ void __launch_bounds__(NT)
qrmse_finalize(const float* __restrict__ partials, int nblocks, float* __restrict__ out) {
    // accumulate block partials in double (keeps ~35M-scale counts exact)
    double acc[10];
#pragma unroll
    for (int j = 0; j < 10; ++j) acc[j] = 0.0;
    for (int i = threadIdx.x; i < nblocks; i += NT) {
#pragma unroll
        for (int j = 0; j < 10; ++j) acc[j] += (double)partials[i * 10 + j];
    }

    __shared__ double sd[NT];
    __shared__ double tot[10];
    for (int j = 0; j < 10; ++j) {
        sd[threadIdx.x] = acc[j];
        __syncthreads();
        for (int off = NT / 2; off > 0; off >>= 1) {
            if (threadIdx.x < off) sd[threadIdx.x] += sd[threadIdx.x + off];
            __syncthreads();
        }
        if (threadIdx.x == 0) tot[j] = sd[0];
        __syncthreads();
    }

    // every thread computes the per-bin statistics (keeps EXEC uniform)
    float S[5], C[5];
#pragma unroll
    for (int j = 0; j < 5; ++j) {
        S[j] = (float)tot[j];
        C[j] = (float)tot[5 + j];
    }

    float mseb[4], wnb[4], wb[4];
    float wsum = 0.f, anyp = 0.f;
#pragma unroll
    for (int b = 0; b < 4; ++b) {
        float cnt     = C[b] - C[b + 1];
        float ssq     = S[b] - S[b + 1];
        float present = (cnt > 0.f) ? 1.0f : 0.0f;
        anyp          = fmaxf(anyp, present);
        float csafe   = (cnt > 0.f) ? cnt : 1.0f;
        mseb[b]       = ssq / csafe;
        wb[b]         = (cnt > 0.f) ? (1.0f / csafe) : 0.0f;
        wsum         += wb[b];
    }
    float wden = (wsum > 0.f) ? wsum : 1.0f;
#pragma unroll
    for (int b = 0; b < 4; ++b) wnb[b] = wb[b] / wden;

    // weighted_mse = sum_b wn[b]*mse[b] as a K=4 dot on the matrix unit.
    // A (16x4 f32): lane0 vgpr{0,1} = A[0][{0,1}], lane16 vgpr{0,1} = A[0][{2,3}]
    // B (4x16 f32): lane0 vgpr{0,1} = B[{0,1}][0], lane16 vgpr{0,1} = B[{2,3}][0]
    // D[0][0] lives in lane 0, VGPR 0.
    const int lane = threadIdx.x & 31;
    const float l0  = (lane == 0)  ? 1.0f : 0.0f;
    const float l16 = (lane == 16) ? 1.0f : 0.0f;
    v2f A, B;
    A.x = l0 * wnb[0]  + l16 * wnb[2];
    A.y = l0 * wnb[1]  + l16 * wnb[3];
    B.x = l0 * mseb[0] + l16 * mseb[2];
    B.y = l0 * mseb[1] + l16 * mseb[3];
    v8f Cm = {0.f, 0.f, 0.f, 0.f, 0.f, 0.f, 0.f, 0.f};

    float wmse;
#if __has_builtin(__builtin_amdgcn_wmma_f32_16x16x4_f32)
    v8f D = __builtin_amdgcn_wmma_f32_16x16x4_f32(
        /*neg_a=*/false, A, /*neg_b=*/false, B,
        /*c_mod=*/(short)0, Cm, /*reuse_a=*/false, /*reuse_b=*/false);
    wmse = D[0];
#else
    wmse = wnb[0] * mseb[0] + wnb[1] * mseb[1] + wnb[2] * mseb[2] + wnb[3] * mseb[3];
#endif

    if (threadIdx.x == 0) {
        float loss = sqrtf(wmse + 1e-8f);
        out[0] = (anyp > 0.f) ? loss : 0.0f;
    }
}

extern "C" void kernel_launch(void* const* d_in, const int* in_sizes, int n_in,
                              void* d_out, int out_size, void* d_ws, size_t ws_size,
                              hipStream_t stream) {
    const float* yp = (const float*)d_in[0];   // y_pred, f32
    const float* yt = (const float*)d_in[1];   // y_true, f32
    float* out      = (float*)d_out;           // scalar f32 loss
    float* partials = (float*)d_ws;            // NB * 10 floats of scratch
    const int n = in_sizes[0];

    qrmse_partial<<<NB, NT, 0, stream>>>(yp, yt, partials, n);
    qrmse_finalize<<<1, NT, 0, stream>>>(partials, NB, out);
}